// deformableResBlock_74586402063012
// MI455X (gfx1250) — compile-verified
//
#include <hip/hip_runtime.h>

// Problem constants (B=1, C=32, D=24, H=96, W=96, G=4)
#define DD   24
#define HH   96
#define WW   96
#define HW   (HH*WW)                 // 9216
#define DHW  (DD*HH*WW)              // 221184
#define NGRP_ELEMS (8*DHW)           // elements per group (C/G=8)
#define NTILES (DD*HH*(WW/16))       // 13824 wave tiles
#define AP_ELEMS (4*216*64)          // packed offset-conv weights: 4 mtiles x 216 ksteps x 64
#define DP_ELEMS (27*2*8*64)         // packed deform weights: 27 taps x 2 mtiles x 8 ksteps x 64

typedef __attribute__((ext_vector_type(2))) float v2f;
typedef __attribute__((ext_vector_type(8))) float v8f;

__device__ __forceinline__ v8f wmma_f32(v2f a, v2f b, v8f c) {
  // D = A(16x4) * B(4x16) + C(16x16), fp32, wave32
  return __builtin_amdgcn_wmma_f32_16x16x4_f32(false, a, false, b, (short)0, c, false, false);
}

// ---------------------------------------------------------------------------
// Weight packing into WMMA A-panel layout:
// panel[((mt*216)+kst)*64 + lane*2 + j] = W[o = mt*16 + lane%16][K = kst*4 + (lane/16)*2 + j]
// K = tap*32 + c, tap = di*9 + dj*3 + dk  (cross-correlation, matches jax conv)
// ---------------------------------------------------------------------------
__global__ void pack_ow_kernel(const float* __restrict__ ow, float* __restrict__ ap) {
  int idx = blockIdx.x * blockDim.x + threadIdx.x;
  if (idx >= AP_ELEMS) return;
  int pos  = idx & 63;
  int blk  = idx >> 6;
  int lane = pos >> 1;
  int j    = pos & 1;
  int mt   = blk / 216;
  int kst  = blk % 216;
  int o    = mt * 16 + (lane & 15);
  int K    = kst * 4 + (lane >> 4) * 2 + j;
  int tap  = K >> 5;
  int c    = K & 31;
  int di = tap / 9, dj = (tap / 3) % 3, dk = tap % 3;
  float v = 0.0f;
  if (o < 54) v = ow[((((size_t)o * 32 + c) * 3 + di) * 3 + dj) * 3 + dk];
  ap[idx] = v;
}

__global__ void pack_dw_kernel(const float* __restrict__ dw, float* __restrict__ dp) {
  int idx = blockIdx.x * blockDim.x + threadIdx.x;
  if (idx >= DP_ELEMS) return;
  int pos  = idx & 63;
  int blk  = idx >> 6;          // tap*16 + mt*8 + ks
  int lane = pos >> 1;
  int j    = pos & 1;
  int tap  = blk >> 4;
  int mt   = (blk >> 3) & 1;
  int ks   = blk & 7;
  int o    = mt * 16 + (lane & 15);
  int c    = ks * 4 + (lane >> 4) * 2 + j;
  int di = tap / 9, dj = (tap / 3) % 3, dk = tap % 3;
  dp[idx] = dw[((((size_t)o * 32 + c) * 3 + di) * 3 + dj) * 3 + dk];
}

// ---------------------------------------------------------------------------
// GroupNorm statistics: block-reduce + atomics into stats[g] (sum) / stats[4+g] (sumsq)
// layout 0: NCDHW (group = contiguous 8*DHW chunk);  layout 1: DHWC
// ---------------------------------------------------------------------------
__global__ void gn_stats(const float* __restrict__ in, float* __restrict__ stats, int layout) {
  int g   = blockIdx.y;
  int tid = threadIdx.x;
  float s = 0.0f, s2 = 0.0f;
  for (int e = blockIdx.x * blockDim.x + tid; e < NGRP_ELEMS; e += gridDim.x * blockDim.x) {
    float v;
    if (layout == 0) {
      v = in[(size_t)g * NGRP_ELEMS + e];
    } else {
      int vox = e >> 3;
      int c   = (g << 3) + (e & 7);
      v = in[(size_t)vox * 32 + c];
    }
    s += v; s2 += v * v;
  }
  __shared__ float rs[256], rq[256];
  rs[tid] = s; rq[tid] = s2;
  __syncthreads();
  for (int o = 128; o > 0; o >>= 1) {
    if (tid < o) { rs[tid] += rs[tid + o]; rq[tid] += rq[tid + o]; }
    __syncthreads();
  }
  if (tid == 0) {
    atomicAdd(&stats[g], rs[0]);
    atomicAdd(&stats[4 + g], rq[0]);
  }
}

// normalize + relu; transpose_in=1: input NCDHW -> output DHWC; 0: DHWC -> DHWC
__global__ void gn_norm_relu(const float* __restrict__ in, const float* __restrict__ stats,
                             const float* __restrict__ gamma, const float* __restrict__ beta,
                             float* __restrict__ out, int transpose_in) {
  int idx = blockIdx.x * blockDim.x + threadIdx.x;
  if (idx >= 32 * DHW) return;
  int c, vox;
  if (transpose_in) { c = idx / DHW; vox = idx - c * DHW; }
  else              { vox = idx >> 5; c = idx & 31; }
  int g = c >> 3;
  float n    = (float)NGRP_ELEMS;
  float mean = stats[g] / n;
  float var  = stats[4 + g] / n - mean * mean;
  float rsq  = rsqrtf(var + 1e-5f);
  float v = in[idx];
  float y = fmaxf((v - mean) * rsq * gamma[c] + beta[c], 0.0f);
  if (transpose_in) out[(size_t)vox * 32 + c] = y;
  else              out[idx] = y;
}

// ---------------------------------------------------------------------------
// Offset conv: implicit GEMM [64pad x 864] x [864 x 16vox] via WMMA f32 16x16x4.
// One wave per 16-voxel tile along W. Output channel-last [vox][64].
// ---------------------------------------------------------------------------
__global__ __launch_bounds__(128)
void offset_conv_wmma(const float* __restrict__ hin, const float* __restrict__ apanel,
                      const float* __restrict__ ob, float* __restrict__ offb) {
  int wave = blockIdx.x * 4 + (threadIdx.x >> 5);
  int lane = threadIdx.x & 31;
  if (wave >= NTILES) return;
  int tw = wave % 6;
  int th = (wave / 6) % HH;
  int td = wave / (6 * HH);
  int w0 = tw * 16;
  int ln   = lane & 15;   // voxel within tile
  int lhid = lane >> 4;   // K-pair half

  v8f acc0 = {}, acc1 = {}, acc2 = {}, acc3 = {};

  for (int di = 0; di < 3; ++di) {
    int dd = td + di - 1;
    if ((unsigned)dd >= (unsigned)DD) continue;            // zero-pad D (wave-uniform)
    for (int dj = 0; dj < 3; ++dj) {
      int hh = th + dj - 1;
      if ((unsigned)hh >= (unsigned)HH) continue;          // zero-pad H (wave-uniform)
      const float* plane = hin + ((size_t)dd * HW + (size_t)hh * WW) * 32;
      for (int dk = 0; dk < 3; ++dk) {
        int tap = (di * 3 + dj) * 3 + dk;
        int wn  = w0 + ln + dk - 1;
        float m = ((unsigned)wn < (unsigned)WW) ? 1.0f : 0.0f;  // per-lane W pad mask
        int wc  = wn < 0 ? 0 : (wn > WW - 1 ? WW - 1 : wn);
        const float* src = plane + (size_t)wc * 32 + lhid * 2;
        const float* ap  = apanel + (size_t)(tap * 8) * 64 + lane * 2;
#pragma unroll
        for (int ks = 0; ks < 8; ++ks) {
          v2f b = *(const v2f*)(src + ks * 4);
          b *= m;
          v2f a0 = *(const v2f*)(ap + (0 * 216 + ks) * 64);
          acc0 = wmma_f32(a0, b, acc0);
          v2f a1 = *(const v2f*)(ap + (1 * 216 + ks) * 64);
          acc1 = wmma_f32(a1, b, acc1);
          v2f a2 = *(const v2f*)(ap + (2 * 216 + ks) * 64);
          acc2 = wmma_f32(a2, b, acc2);
          v2f a3 = *(const v2f*)(ap + (3 * 216 + ks) * 64);
          acc3 = wmma_f32(a3, b, acc3);
        }
      }
    }
  }

  size_t vox = (size_t)td * HW + (size_t)th * WW + (w0 + ln);
  float* op = offb + vox * 64;
#pragma unroll
  for (int r = 0; r < 8; ++r) {
    int o0 = lhid * 8 + r;         op[o0] = acc0[r] + ob[o0];
    int o1 = 16 + lhid * 8 + r;    op[o1] = acc1[r] + ob[o1];
    int o2 = 32 + lhid * 8 + r;    op[o2] = acc2[r] + ob[o2];
    int o3 = 48 + lhid * 8 + r;
    if (o3 < 54) op[o3] = acc3[r] + ob[o3];
  }
}

// ---------------------------------------------------------------------------
// Deformable conv: per tap, bilinear-gather B panel (4 masked float2 loads per
// K-step) then WMMA accumulate [32 x 32] x [32 x 16vox].
// nchw_out=0: write DHWC (+bias). nchw_out=1: write NCDHW (+bias +residual x).
// ---------------------------------------------------------------------------
__global__ __launch_bounds__(128)
void deform_conv_wmma(const float* __restrict__ hin, const float* __restrict__ offb,
                      const float* __restrict__ dwpanel, const float* __restrict__ db,
                      const float* __restrict__ resid, float* __restrict__ out, int nchw_out) {
  int wave = blockIdx.x * 4 + (threadIdx.x >> 5);
  int lane = threadIdx.x & 31;
  if (wave >= NTILES) return;
  int tw = wave % 6;
  int th = (wave / 6) % HH;
  int td = wave / (6 * HH);
  int ln   = lane & 15;
  int lhid = lane >> 4;
  int wv = tw * 16 + ln;
  size_t vox = (size_t)td * HW + (size_t)th * WW + wv;
  const float* offp = offb + vox * 64;

  v8f acc0 = {}, acc1 = {};

  for (int i = 0; i < 3; ++i) {
    int dd = td + i - 1;
    if ((unsigned)dd >= (unsigned)DD) continue;            // D zero-pad (wave-uniform)
    const float* plane = hin + (size_t)dd * HW * 32;
    for (int j = 0; j < 3; ++j) {
      for (int k = 0; k < 3; ++k) {
        int tap = (i * 3 + j) * 3 + k;
        v2f off2 = *(const v2f*)(offp + tap * 2);
        float hp = (float)(th + j - 1) + off2[0];
        float wp = (float)(wv + k - 1) + off2[1];
        float h0f = floorf(hp), w0f = floorf(wp);
        int h0i = (int)h0f, w0i = (int)w0f;
        float fh = hp - h0f, fw = wp - w0f;
        float m00 = ((unsigned)h0i       < (unsigned)HH && (unsigned)w0i       < (unsigned)WW) ? 1.f : 0.f;
        float m01 = ((unsigned)h0i       < (unsigned)HH && (unsigned)(w0i + 1) < (unsigned)WW) ? 1.f : 0.f;
        float m10 = ((unsigned)(h0i + 1) < (unsigned)HH && (unsigned)w0i       < (unsigned)WW) ? 1.f : 0.f;
        float m11 = ((unsigned)(h0i + 1) < (unsigned)HH && (unsigned)(w0i + 1) < (unsigned)WW) ? 1.f : 0.f;
        float wt00 = (1.f - fh) * (1.f - fw) * m00;
        float wt01 = (1.f - fh) * fw * m01;
        float wt10 = fh * (1.f - fw) * m10;
        float wt11 = fh * fw * m11;
        int h0c = h0i < 0 ? 0 : (h0i > HH - 1 ? HH - 1 : h0i);
        int h1i = h0i + 1;
        int h1c = h1i < 0 ? 0 : (h1i > HH - 1 ? HH - 1 : h1i);
        int w0c = w0i < 0 ? 0 : (w0i > WW - 1 ? WW - 1 : w0i);
        int w1i = w0i + 1;
        int w1c = w1i < 0 ? 0 : (w1i > WW - 1 ? WW - 1 : w1i);
        const float* p00 = plane + ((size_t)h0c * WW + w0c) * 32 + lhid * 2;
        const float* p01 = plane + ((size_t)h0c * WW + w1c) * 32 + lhid * 2;
        const float* p10 = plane + ((size_t)h1c * WW + w0c) * 32 + lhid * 2;
        const float* p11 = plane + ((size_t)h1c * WW + w1c) * 32 + lhid * 2;
        const float* ap  = dwpanel + (size_t)(tap * 16) * 64 + lane * 2;
#pragma unroll
        for (int ks = 0; ks < 8; ++ks) {
          int c0 = ks * 4;
          v2f b = wt00 * *(const v2f*)(p00 + c0)
                + wt01 * *(const v2f*)(p01 + c0)
                + wt10 * *(const v2f*)(p10 + c0)
                + wt11 * *(const v2f*)(p11 + c0);
          v2f a0 = *(const v2f*)(ap + ks * 64);
          acc0 = wmma_f32(a0, b, acc0);
          v2f a1 = *(const v2f*)(ap + (8 + ks) * 64);
          acc1 = wmma_f32(a1, b, acc1);
        }
      }
    }
  }

  if (nchw_out) {
#pragma unroll
    for (int r = 0; r < 8; ++r) {
      int o0 = lhid * 8 + r;
      out[(size_t)o0 * DHW + vox] = acc0[r] + db[o0] + resid[(size_t)o0 * DHW + vox];
      int o1 = 16 + lhid * 8 + r;
      out[(size_t)o1 * DHW + vox] = acc1[r] + db[o1] + resid[(size_t)o1 * DHW + vox];
    }
  } else {
    float* op = out + vox * 32;
#pragma unroll
    for (int r = 0; r < 8; ++r) {
      int o0 = lhid * 8 + r;      op[o0] = acc0[r] + db[o0];
      int o1 = 16 + lhid * 8 + r; op[o1] = acc1[r] + db[o1];
    }
  }
}

// ---------------------------------------------------------------------------
extern "C" void kernel_launch(void* const* d_in, const int* in_sizes, int n_in,
                              void* d_out, int out_size, void* d_ws, size_t ws_size,
                              hipStream_t stream) {
  (void)in_sizes; (void)n_in; (void)out_size; (void)ws_size;
  const float* x   = (const float*)d_in[0];
  const float* g1  = (const float*)d_in[1];
  const float* be1 = (const float*)d_in[2];
  const float* g2  = (const float*)d_in[3];
  const float* be2 = (const float*)d_in[4];
  const float* ow1 = (const float*)d_in[5];
  const float* ob1 = (const float*)d_in[6];
  const float* dw1 = (const float*)d_in[7];
  const float* db1 = (const float*)d_in[8];
  const float* ow2 = (const float*)d_in[9];
  const float* ob2 = (const float*)d_in[10];
  const float* dw2 = (const float*)d_in[11];
  const float* db2 = (const float*)d_in[12];
  float* out = (float*)d_out;

  // Workspace layout (floats). Total ~113.9 MB; working set stays in the 192MB L2.
  float* w     = (float*)d_ws;
  float* stats = w;                      // 16 (stats for GN1 at [0..7], GN2 at [8..15])
  float* ap1   = w + 64;                 // AP_ELEMS
  float* ap2   = ap1 + AP_ELEMS;
  float* dp1   = ap2 + AP_ELEMS;         // DP_ELEMS
  float* dp2   = dp1 + DP_ELEMS;
  float* h1    = dp2 + DP_ELEMS;         // 32*DHW floats, DHWC (reused as h3)
  float* offb  = h1 + (size_t)32 * DHW;  // 64*DHW floats, [vox][64]
  float* h2    = offb + (size_t)64 * DHW;// 32*DHW floats, DHWC

  hipMemsetAsync(stats, 0, 16 * sizeof(float), stream);

  pack_ow_kernel<<<(AP_ELEMS + 255) / 256, 256, 0, stream>>>(ow1, ap1);
  pack_ow_kernel<<<(AP_ELEMS + 255) / 256, 256, 0, stream>>>(ow2, ap2);
  pack_dw_kernel<<<(DP_ELEMS + 255) / 256, 256, 0, stream>>>(dw1, dp1);
  pack_dw_kernel<<<(DP_ELEMS + 255) / 256, 256, 0, stream>>>(dw2, dp2);

  // Stage 1: GN1 + relu (NCDHW -> DHWC)
  gn_stats<<<dim3(256, 4), 256, 0, stream>>>(x, stats, 0);
  gn_norm_relu<<<(32 * DHW + 255) / 256, 256, 0, stream>>>(x, stats, g1, be1, h1, 1);
  // offsets1 + deform1 -> h2 (DHWC)
  offset_conv_wmma<<<NTILES / 4, 128, 0, stream>>>(h1, ap1, ob1, offb);
  deform_conv_wmma<<<NTILES / 4, 128, 0, stream>>>(h1, offb, dp1, db1, nullptr, h2, 0);

  // Stage 2: GN2 + relu (DHWC -> DHWC, h3 aliases h1)
  gn_stats<<<dim3(256, 4), 256, 0, stream>>>(h2, stats + 8, 1);
  gn_norm_relu<<<(32 * DHW + 255) / 256, 256, 0, stream>>>(h2, stats + 8, g2, be2, h1, 0);
  // offsets2 + deform2 (+bias +residual) -> d_out (NCDHW)
  offset_conv_wmma<<<NTILES / 4, 128, 0, stream>>>(h1, ap2, ob2, offb);
  deform_conv_wmma<<<NTILES / 4, 128, 0, stream>>>(h1, offb, dp2, db2, x, out, 1);
}